// WeightedTensorProductBlock_10144712753191
// MI455X (gfx1250) — compile-verified
//
#include <hip/hip_runtime.h>
#include <cstdint>

// ---------------------------------------------------------------------------
// out[b,i,j,f] = 0.75 * w[f] * a[0] * x[b,i,f] * y[b,j,f]
// B = 200000, F = 64, i,j in [0,3). Pure streaming op (~768 MB / call,
// 0.45 FLOP/byte) -> memory bound; no contraction dim -> WMMA inapplicable.
// Strategy: ASYNCcnt double-buffered GLOBAL_LOAD_ASYNC_TO_LDS_B128 staging of
// x/y tiles, ds_load_b128 tile reads, non-temporal global_store_b128 output.
// ---------------------------------------------------------------------------

typedef __attribute__((ext_vector_type(4))) float v4f;
typedef __attribute__((ext_vector_type(4))) int   v4i;
typedef __attribute__((address_space(1))) v4i     gas_v4i;   // global AS
typedef __attribute__((address_space(3))) v4i     las_v4i;   // LDS AS

#define NORM     0.75f
#define TB       16                 // atoms (b) per chunk
#define TILE_V4  (TB * 3 * 16)      // 768 float4 per tensor tile (12 KB)
#define THREADS  256                // 8 wave32
#define MAXGRID  1250

#if __has_builtin(__builtin_amdgcn_global_load_async_to_lds_b128) && \
    __has_builtin(__builtin_amdgcn_s_wait_asynccnt)
#define USE_ASYNC_LDS 1
#else
#define USE_ASYNC_LDS 0
#endif

__device__ __forceinline__ void stage16(const v4f* g, v4f* l) {
#if USE_ASYNC_LDS
  // per-lane 16B DMA: global -> LDS, tracked by ASYNCcnt
  v4i* gg = (v4i*)g;   // generic reinterpret (drops const)
  v4i* ll = (v4i*)l;
  __builtin_amdgcn_global_load_async_to_lds_b128(
      (gas_v4i*)gg, (las_v4i*)ll, /*imm offset*/ 0, /*cpol*/ 0);
#else
  *l = *g;   // synchronous fallback (global_load_b128 + ds_store_b128)
#endif
}

__device__ __forceinline__ void wait_stage() {
#if USE_ASYNC_LDS
  __builtin_amdgcn_s_wait_asynccnt(0);
#endif
}

__global__ __launch_bounds__(THREADS)
void wtp_main_kernel(const float* __restrict__ w, const float* __restrict__ a,
                     const float* __restrict__ x, const float* __restrict__ y,
                     float* __restrict__ out, int NC, int grid) {
  __shared__ v4f sx[2][TILE_V4];   // 2 x 12 KB
  __shared__ v4f sy[2][TILE_V4];   // 2 x 12 KB  (48 KB total)

  const int t    = threadIdx.x;
  const int bsub = t >> 4;         // 0..15 : atom within chunk
  const int f4   = t & 15;         // 0..15 : float4 channel block

  const v4f* x4   = (const v4f*)x;
  const v4f* y4   = (const v4f*)y;
  v4f*       out4 = (v4f*)out;

  const float s = a[0] * NORM;
  const v4f  wa = ((const v4f*)w)[f4] * s;   // per-channel scale, loaded once

  int c = blockIdx.x;

  // prologue: stage first chunk into buffer 0
  if (c < NC) {
    const long gb = (long)c * TILE_V4;
#pragma unroll
    for (int k = 0; k < 3; ++k) {
      stage16(x4 + gb + k * THREADS + t, &sx[0][k * THREADS + t]);
      stage16(y4 + gb + k * THREADS + t, &sy[0][k * THREADS + t]);
    }
  }

  int p = 0;
  for (; c < NC; c += grid, p ^= 1) {
    wait_stage();        // this wave's async loads for buf p are in LDS
    __syncthreads();     // ... and everyone else's too

    // kick off next chunk into the other buffer (fully consumed last iter)
    const int cn = c + grid;
    if (cn < NC) {
      const long gb = (long)cn * TILE_V4;
#pragma unroll
      for (int k = 0; k < 3; ++k) {
        stage16(x4 + gb + k * THREADS + t, &sx[p ^ 1][k * THREADS + t]);
        stage16(y4 + gb + k * THREADS + t, &sy[p ^ 1][k * THREADS + t]);
      }
    }

    // tile layout: v4 index (bsub*3 + i)*16 + f4
    v4f xv[3], yv[3];
#pragma unroll
    for (int i = 0; i < 3; ++i) {
      xv[i] = sx[p][(bsub * 3 + i) * 16 + f4];
      yv[i] = sy[p][(bsub * 3 + i) * 16 + f4];
    }

    const long b = (long)c * TB + bsub;
    v4f* o = out4 + b * 144 + f4;            // 144 v4f per atom
#pragma unroll
    for (int i = 0; i < 3; ++i) {
      const v4f wx = wa * xv[i];
#pragma unroll
      for (int j = 0; j < 3; ++j) {
        // write-once 461 MB stream: keep it out of the caches
        __builtin_nontemporal_store(wx * yv[j], o + (i * 3 + j) * 16);
      }
    }
    __syncthreads();     // buf p free for reuse
  }
}

// tail for B % 16 != 0 (unused for B = 200000, kept for generality)
__global__ __launch_bounds__(THREADS)
void wtp_tail_kernel(const float* __restrict__ w, const float* __restrict__ a,
                     const float* __restrict__ x, const float* __restrict__ y,
                     float* __restrict__ out, int b0, int Bn) {
  const int t = threadIdx.x;
  const int b = b0 + (t >> 4);
  if (b >= Bn) return;
  const int f4 = t & 15;

  const v4f* x4   = (const v4f*)x;
  const v4f* y4   = (const v4f*)y;
  v4f*       out4 = (v4f*)out;

  const v4f wa = ((const v4f*)w)[f4] * (a[0] * NORM);
  v4f xv[3], yv[3];
#pragma unroll
  for (int i = 0; i < 3; ++i) {
    xv[i] = x4[((long)b * 3 + i) * 16 + f4];
    yv[i] = y4[((long)b * 3 + i) * 16 + f4];
  }
  v4f* o = out4 + (long)b * 144 + f4;
#pragma unroll
  for (int i = 0; i < 3; ++i) {
    const v4f wx = wa * xv[i];
#pragma unroll
    for (int j = 0; j < 3; ++j)
      __builtin_nontemporal_store(wx * yv[j], o + (i * 3 + j) * 16);
  }
}

extern "C" void kernel_launch(void* const* d_in, const int* in_sizes, int n_in,
                              void* d_out, int out_size, void* d_ws, size_t ws_size,
                              hipStream_t stream) {
  const float* w = (const float*)d_in[0];   // [F]
  const float* a = (const float*)d_in[1];   // [1]
  const float* x = (const float*)d_in[2];   // [B,3,F]
  const float* y = (const float*)d_in[3];   // [B,3,F]
  float* out = (float*)d_out;               // [B,3,3,F]

  const int Bn = in_sizes[2] / 192;         // B (3*64 floats per atom)
  const int NC = Bn / TB;                   // full 16-atom chunks

  if (NC > 0) {
    const int grid = NC < MAXGRID ? NC : MAXGRID;
    wtp_main_kernel<<<grid, THREADS, 0, stream>>>(w, a, x, y, out, NC, grid);
  }
  if (Bn - NC * TB > 0) {
    wtp_tail_kernel<<<1, THREADS, 0, stream>>>(w, a, x, y, out, NC * TB, Bn);
  }
}